// ImportanceRenderer_41686952575365
// MI455X (gfx1250) — compile-verified
//
#include <hip/hip_runtime.h>
#include <math.h>

typedef float v2f __attribute__((ext_vector_type(2)));
typedef float v8f __attribute__((ext_vector_type(8)));

#define RESX   64
#define R_RAYS 4096
#define NS     48
#define NI     48
#define NCS    196608       // R_RAYS * NS
#define FOCAL  225.0f
#define FH     1024
#define FW     1024

// workspace layout (float offsets)
static constexpr int OFF_Z      = 0;
static constexpr int OFF_CX     = OFF_Z     + NCS;
static constexpr int OFF_CY     = OFF_CX    + NCS;
static constexpr int OFF_DENSC  = OFF_CY    + NCS;
static constexpr int OFF_COLC   = OFF_DENSC + NCS;
static constexpr int OFF_WC     = OFF_COLC  + NCS * 32;
static constexpr int OFF_ZF     = OFF_WC    + R_RAYS * 47;
static constexpr int OFF_FX     = OFF_ZF    + NCS;
static constexpr int OFF_FY     = OFF_FX    + NCS;
static constexpr int OFF_DENSF  = OFF_FY    + NCS;
static constexpr int OFF_COLF   = OFF_DENSF + NCS;
static constexpr int OFF_KEYS   = OFF_COLF  + NCS * 32;   // 6 uints
static constexpr int OFF_PARAMS = OFF_KEYS  + 8;          // 8 floats

__device__ __forceinline__ unsigned f2key(float f) {
  unsigned b = __float_as_uint(f);
  return (b & 0x80000000u) ? ~b : (b | 0x80000000u);
}
__device__ __forceinline__ float key2f(unsigned k) {
  unsigned b = (k & 0x80000000u) ? (k ^ 0x80000000u) : ~k;
  return __uint_as_float(b);
}
__device__ __forceinline__ float softplus1(float x) {
  return fmaxf(x, 0.0f) + log1pf(expf(-fabsf(x)));
}
__device__ __forceinline__ unsigned umn(unsigned a, unsigned b) { return a < b ? a : b; }
__device__ __forceinline__ unsigned umx(unsigned a, unsigned b) { return a > b ? a : b; }

// ---------------- init atomic slots ----------------
__global__ void k_init(float* ws) {
  unsigned* keys = (unsigned*)(ws + OFF_KEYS);
  if (threadIdx.x == 0) {
    keys[0] = 0xFFFFFFFFu; keys[1] = 0u;   // x min/max
    keys[2] = 0xFFFFFFFFu; keys[3] = 0u;   // y min/max
    keys[4] = 0xFFFFFFFFu; keys[5] = 0u;   // depth min/max
  }
}

// ---------------- rays, stratified z, coarse coords, min/max ----------------
__global__ __launch_bounds__(256) void k_setup(const float* __restrict__ t_rand,
                                               float* __restrict__ ws) {
  int idx = blockIdx.x * 256 + threadIdx.x;            // < NCS exactly
  int r = idx / NS, s = idx % NS;
  int col = r % RESX, row = r / RESX;
  float dirx = (col - 32.0f) / FOCAL;
  float diry = -(row - 32.0f) / FOCAL;
  float ts = s * (1.0f / 47.0f);
  float lower = (s == 0)  ? 0.0f : 0.5f * ((s - 1) * (1.0f / 47.0f) + ts);
  float upper = (s == 47) ? 1.0f : 0.5f * (ts + (s + 1) * (1.0f / 47.0f));
  float z = lower + (upper - lower) * t_rand[idx];
  float x = dirx * z, y = diry * z;
  ws[OFF_Z  + idx] = z;
  ws[OFF_CX + idx] = x;
  ws[OFF_CY + idx] = y;

  unsigned kxm = f2key(x), kxM = kxm, kym = f2key(y), kyM = kym, kzm = f2key(z), kzM = kzm;
  #pragma unroll
  for (int o = 16; o > 0; o >>= 1) {
    kxm = umn(kxm, (unsigned)__shfl_xor((int)kxm, o, 32));
    kxM = umx(kxM, (unsigned)__shfl_xor((int)kxM, o, 32));
    kym = umn(kym, (unsigned)__shfl_xor((int)kym, o, 32));
    kyM = umx(kyM, (unsigned)__shfl_xor((int)kyM, o, 32));
    kzm = umn(kzm, (unsigned)__shfl_xor((int)kzm, o, 32));
    kzM = umx(kzM, (unsigned)__shfl_xor((int)kzM, o, 32));
  }
  if ((threadIdx.x & 31) == 0) {
    unsigned* keys = (unsigned*)(ws + OFF_KEYS);
    atomicMin(&keys[0], kxm); atomicMax(&keys[1], kxM);
    atomicMin(&keys[2], kym); atomicMax(&keys[3], kyM);
    atomicMin(&keys[4], kzm); atomicMax(&keys[5], kzM);
  }
}

// ---------------- normalization params ----------------
__global__ void k_params(float* ws) {
  if (threadIdx.x != 0) return;
  unsigned* keys = (unsigned*)(ws + OFF_KEYS);
  float xmin = key2f(keys[0]), xmax = key2f(keys[1]);
  float ymin = key2f(keys[2]), ymax = key2f(keys[3]);
  float* p = ws + OFF_PARAMS;
  // coarse: xn = 2(x-xmin)/(xmax-xmin)-1 ; px = (xn+1)*W/2 - 0.5 = x*sx+ox
  float sx = (float)FW / (xmax - xmin);
  float sy = (float)FH / (ymax - ymin);
  p[0] = sx;            p[1] = -xmin * sx - 0.5f;
  p[2] = sy;            p[3] = -ymin * sy - 0.5f;
  // fine: raw grid coords: px = (x+1)*W/2 - 0.5
  p[4] = 0.5f * FW;     p[5] = 0.5f * FW - 0.5f;
  p[6] = 0.5f * FH;     p[7] = 0.5f * FH - 0.5f;
}

// ---------------- grid_sample + MLP decode (WMMA) ----------------
// 128 threads = 4 waves; each wave computes a 16-sample tile.
__global__ __launch_bounds__(128) void k_decode(
    const float* __restrict__ fm, const float* __restrict__ W1,
    const float* __restrict__ b1, const float* __restrict__ W2,
    const float* __restrict__ b2, const float* __restrict__ gx,
    const float* __restrict__ gy, const float* __restrict__ prm,
    float* __restrict__ colOut, float* __restrict__ denOut) {
  __shared__ float w1s[32][65];
  __shared__ float w2s[64][49];
  __shared__ float b1s[64];
  __shared__ float b2s[48];
  __shared__ float fs[4][16][33];
  __shared__ float hs[4][16][65];

  int tid = threadIdx.x;
  int wid = tid >> 5, lane = tid & 31;

  for (int i = tid; i < 32 * 64; i += 128) w1s[i >> 6][i & 63] = W1[i];
  for (int i = tid; i < 64 * 48; i += 128) {
    int k = i / 48, n = i % 48;
    w2s[k][n] = (n < 33) ? W2[k * 33 + n] : 0.0f;
  }
  for (int i = tid; i < 64; i += 128) b1s[i] = b1[i];
  for (int i = tid; i < 48; i += 128) b2s[i] = (i < 33) ? b2[i] : 0.0f;

  int base  = (blockIdx.x * 4 + wid) * 16;
  int sLoc  = lane & 15;
  int half  = lane >> 4;
  int koff  = half * 2;
  int rbase = half * 8;

  // -------- bilinear gather: lane handles (sample = sLoc, channels half*16..+15)
  {
    float sx = prm[0], ox = prm[1], sy = prm[2], oy = prm[3];
    float X = gx[base + sLoc] * sx + ox;
    float Y = gy[base + sLoc] * sy + oy;
    float x0 = floorf(X), y0 = floorf(Y);
    float wx = X - x0, wy = Y - y0;
    float x1 = x0 + 1.0f, y1 = y0 + 1.0f;
    float vx0 = (x0 >= 0.0f && x0 <= (float)(FW - 1)) ? 1.0f : 0.0f;
    float vx1 = (x1 >= 0.0f && x1 <= (float)(FW - 1)) ? 1.0f : 0.0f;
    float vy0 = (y0 >= 0.0f && y0 <= (float)(FH - 1)) ? 1.0f : 0.0f;
    float vy1 = (y1 >= 0.0f && y1 <= (float)(FH - 1)) ? 1.0f : 0.0f;
    float w00 = (1.0f - wx) * (1.0f - wy) * vx0 * vy0;
    float w10 = wx * (1.0f - wy) * vx1 * vy0;
    float w01 = (1.0f - wx) * wy * vx0 * vy1;
    float w11 = wx * wy * vx1 * vy1;
    int ix0 = (int)fminf(fmaxf(x0, 0.0f), (float)(FW - 1));
    int ix1 = (int)fminf(fmaxf(x1, 0.0f), (float)(FW - 1));
    int iy0 = (int)fminf(fmaxf(y0, 0.0f), (float)(FH - 1));
    int iy1 = (int)fminf(fmaxf(y1, 0.0f), (float)(FH - 1));
    int o00 = iy0 * FW + ix0, o10 = iy0 * FW + ix1;
    int o01 = iy1 * FW + ix0, o11 = iy1 * FW + ix1;
    #pragma unroll
    for (int c = 0; c < 16; ++c) {
      const float* f = fm + (((half << 4) + c) << 20);
      float v = w00 * f[o00] + w10 * f[o10] + w01 * f[o01] + w11 * f[o11];
      fs[wid][sLoc][(half << 4) + c] = v;
    }
  }
  __syncthreads();

  // -------- GEMM1: h = softplus(feat(16x32) @ W1(32x64) + b1)
  #pragma unroll
  for (int nt = 0; nt < 4; ++nt) {
    int n0 = nt * 16;
    v8f acc;
    float bb = b1s[n0 + sLoc];
    #pragma unroll
    for (int j = 0; j < 8; ++j) acc[j] = bb;
    #pragma unroll
    for (int k4 = 0; k4 < 8; ++k4) {
      v2f a, b;
      a[0] = fs[wid][sLoc][4 * k4 + koff];
      a[1] = fs[wid][sLoc][4 * k4 + koff + 1];
      b[0] = w1s[4 * k4 + koff][n0 + sLoc];
      b[1] = w1s[4 * k4 + koff + 1][n0 + sLoc];
      acc = __builtin_amdgcn_wmma_f32_16x16x4_f32(false, a, false, b,
                                                  (short)0, acc, false, false);
    }
    #pragma unroll
    for (int j = 0; j < 8; ++j)
      hs[wid][rbase + j][n0 + sLoc] = softplus1(acc[j]);
  }
  __syncthreads();

  // -------- GEMM2: o = h(16x64) @ W2(64x48pad) + b2 ; sigma/rgb postprocess
  #pragma unroll
  for (int nt = 0; nt < 3; ++nt) {
    int n0 = nt * 16;
    v8f acc;
    float bb = b2s[n0 + sLoc];
    #pragma unroll
    for (int j = 0; j < 8; ++j) acc[j] = bb;
    #pragma unroll
    for (int k4 = 0; k4 < 16; ++k4) {
      v2f a, b;
      a[0] = hs[wid][sLoc][4 * k4 + koff];
      a[1] = hs[wid][sLoc][4 * k4 + koff + 1];
      b[0] = w2s[4 * k4 + koff][n0 + sLoc];
      b[1] = w2s[4 * k4 + koff + 1][n0 + sLoc];
      acc = __builtin_amdgcn_wmma_f32_16x16x4_f32(false, a, false, b,
                                                  (short)0, acc, false, false);
    }
    int ct = n0 + sLoc;
    #pragma unroll
    for (int j = 0; j < 8; ++j) {
      int sIdx = base + rbase + j;
      float v = acc[j];
      if (ct == 0) {
        denOut[sIdx] = v;
      } else if (ct < 33) {
        float sg = 1.0f / (1.0f + expf(-v));
        colOut[sIdx * 32 + (ct - 1)] = sg * 1.002f - 0.001f;
      }
    }
  }
}

// ---------------- coarse ray march: weights only ----------------
__global__ __launch_bounds__(256) void k_march_coarse(float* __restrict__ ws) {
  int r = blockIdx.x * 256 + threadIdx.x;
  if (r >= R_RAYS) return;
  const float* z = ws + OFF_Z + r * NS;
  const float* d = ws + OFF_DENSC + r * NS;
  float* w = ws + OFF_WC + r * 47;
  float T = 1.0f, dprev = d[0], zprev = z[0];
  for (int i = 0; i < 47; ++i) {
    float dn = d[i + 1], zn = z[i + 1];
    float smid = softplus1(0.5f * (dprev + dn) - 1.0f);
    float alpha = 1.0f - expf(-smid * (zn - zprev));
    w[i] = alpha * T;
    T *= (1.0f - alpha + 1e-10f);
    dprev = dn; zprev = zn;
  }
}

// ---------------- importance sampling (inverse CDF) + fine coords ----------------
__global__ __launch_bounds__(64) void k_importance(const float* __restrict__ u,
                                                   float* __restrict__ ws) {
  int r = blockIdx.x * 64 + threadIdx.x;
  if (r >= R_RAYS) return;
  float zl[48], wl[47];
  for (int i = 0; i < 48; ++i) zl[i] = ws[OFF_Z + r * NS + i];
  for (int i = 0; i < 47; ++i) wl[i] = ws[OFF_WC + r * 47 + i];
  // smoothed weights: m_i (i=0..47), wsm_i = 0.5*(m_i+m_{i+1})+0.01, take i=1..45
  float m[48];
  m[0] = wl[0]; m[47] = wl[46];
  for (int i = 1; i <= 46; ++i) m[i] = fmaxf(wl[i - 1], wl[i]);
  float w45[45];
  float sum = 0.0f;
  for (int j = 0; j < 45; ++j) {
    w45[j] = 0.5f * (m[j + 1] + m[j + 2]) + 0.01f + 1e-5f;
    sum += w45[j];
  }
  float cdf[46];
  cdf[0] = 0.0f;
  for (int j = 0; j < 45; ++j) cdf[j + 1] = cdf[j] + w45[j] / sum;

  int col = r % RESX, row = r / RESX;
  float dirx = (col - 32.0f) / FOCAL;
  float diry = -(row - 32.0f) / FOCAL;
  unsigned dkm = 0xFFFFFFFFu, dkM = 0u;
  for (int t = 0; t < NI; ++t) {
    float uu = u[r * NI + t];
    int ind = 0;
    for (int j = 0; j < 46; ++j) ind += (cdf[j] <= uu) ? 1 : 0;  // searchsorted right
    int below = ind - 1; if (below < 0) below = 0;
    int above = ind;     if (above > 45) above = 45;
    float cb = cdf[below], ca = cdf[above];
    float bb = 0.5f * (zl[below] + zl[below + 1]);
    float ba = 0.5f * (zl[above] + zl[above + 1]);
    float den = ca - cb; if (den < 1e-5f) den = 1.0f;
    float dpt = bb + (uu - cb) / den * (ba - bb);
    ws[OFF_ZF + r * NI + t] = dpt;
    ws[OFF_FX + r * NI + t] = dirx * dpt;
    ws[OFF_FY + r * NI + t] = diry * dpt;
    unsigned k = f2key(dpt);
    dkm = umn(dkm, k); dkM = umx(dkM, k);
  }
  unsigned* keys = (unsigned*)(ws + OFF_KEYS);
  atomicMin(&keys[4], dkm);
  atomicMax(&keys[5], dkM);
}

// ---------------- merge-sort 96 samples, final ray march ----------------
__global__ __launch_bounds__(32) void k_final(const float* __restrict__ ws,
                                              float* __restrict__ out) {
  __shared__ float dep[96];
  __shared__ float sig[96];
  __shared__ float colS[96][33];
  __shared__ int   ord[96];
  __shared__ float wS[95];
  __shared__ float scal[2];
  int r = blockIdx.x, tid = threadIdx.x;

  for (int t = tid; t < 96; t += 32) {
    float d, s; const float* c;
    if (t < 48) {
      d = ws[OFF_Z + r * NS + t];
      s = ws[OFF_DENSC + r * NS + t];
      c = ws + OFF_COLC + (size_t)(r * NS + t) * 32;
    } else {
      int q = t - 48;
      d = ws[OFF_ZF + r * NI + q];
      s = ws[OFF_DENSF + r * NI + q];
      c = ws + OFF_COLF + (size_t)(r * NI + q) * 32;
    }
    dep[t] = d; sig[t] = s;
    for (int ch = 0; ch < 32; ++ch) colS[t][ch] = c[ch];
  }
  __syncthreads();

  for (int t = tid; t < 96; t += 32) {           // stable rank sort by depth
    float d = dep[t];
    int rank = 0;
    for (int j = 0; j < 96; ++j) {
      float dj = dep[j];
      rank += (dj < d || (dj == d && j < t)) ? 1 : 0;
    }
    ord[rank] = t;
  }
  __syncthreads();

  if (tid == 0) {                                 // sequential transmittance
    float T = 1.0f, wtot = 0.0f, dnum = 0.0f;
    for (int i = 0; i < 95; ++i) {
      int i0 = ord[i], i1 = ord[i + 1];
      float d0 = dep[i0], d1 = dep[i1];
      float smid = softplus1(0.5f * (sig[i0] + sig[i1]) - 1.0f);
      float alpha = 1.0f - expf(-smid * (d1 - d0));
      float w = alpha * T;
      T *= (1.0f - alpha + 1e-10f);
      wS[i] = w;
      wtot += w;
      dnum += w * 0.5f * (d0 + d1);
    }
    scal[0] = wtot; scal[1] = dnum;
  }
  __syncthreads();

  float acc = 0.0f;                               // per-lane color channel
  for (int i = 0; i < 95; ++i) {
    float cm = 0.5f * (colS[ord[i]][tid] + colS[ord[i + 1]][tid]);
    acc += wS[i] * cm;
  }
  out[r * 32 + tid] = acc * 2.0f - 1.0f;

  if (tid == 0) {
    const unsigned* keys = (const unsigned*)(ws + OFF_KEYS);
    float dmin = key2f(keys[4]), dmax = key2f(keys[5]);
    float wtot = scal[0];
    float q = scal[1] / wtot;
    if (q != q) q = __builtin_inff();             // nan -> inf -> clipped to dmax
    q = fminf(fmaxf(q, dmin), dmax);
    out[R_RAYS * 32 + r] = q;
    out[R_RAYS * 33 + r] = wtot;
  }
}

extern "C" void kernel_launch(void* const* d_in, const int* in_sizes, int n_in,
                              void* d_out, int out_size, void* d_ws, size_t ws_size,
                              hipStream_t stream) {
  const float* fm     = (const float*)d_in[0];
  const float* W1     = (const float*)d_in[1];
  const float* b1     = (const float*)d_in[2];
  const float* W2     = (const float*)d_in[3];
  const float* b2     = (const float*)d_in[4];
  const float* t_rand = (const float*)d_in[5];
  const float* u      = (const float*)d_in[6];
  float* out = (float*)d_out;
  float* ws  = (float*)d_ws;
  (void)in_sizes; (void)n_in; (void)out_size; (void)ws_size;

  k_init<<<1, 32, 0, stream>>>(ws);
  k_setup<<<NCS / 256, 256, 0, stream>>>(t_rand, ws);
  k_params<<<1, 32, 0, stream>>>(ws);
  k_decode<<<NCS / 64, 128, 0, stream>>>(fm, W1, b1, W2, b2,
                                         ws + OFF_CX, ws + OFF_CY, ws + OFF_PARAMS,
                                         ws + OFF_COLC, ws + OFF_DENSC);
  k_march_coarse<<<R_RAYS / 256, 256, 0, stream>>>(ws);
  k_importance<<<R_RAYS / 64, 64, 0, stream>>>(u, ws);
  k_decode<<<NCS / 64, 128, 0, stream>>>(fm, W1, b1, W2, b2,
                                         ws + OFF_FX, ws + OFF_FY, ws + OFF_PARAMS + 4,
                                         ws + OFF_COLF, ws + OFF_DENSF);
  k_final<<<R_RAYS, 32, 0, stream>>>(ws, out);
}